// PerRowDataDependentAlibi_11201274708514
// MI455X (gfx1250) — compile-verified
//
#include <hip/hip_runtime.h>
#include <hip/hip_bf16.h>

// PerRowDataDependentAlibi — MI455X (gfx1250, wave32)
//
//  k1: proj = x @ W^T via V_WMMA_F32_16X16X4_F32 (full fp32 precision),
//      scatter into q/k workspace laid out [head][n][8]  (3 MB, L2-resident).
//  k2: fused qk^T -> log_sigmoid -> causal mask -> reverse cumsum -> store.
//      One wave per (head, 16-row band); right-to-left over 256 col-tiles:
//      zero-fill loop (b128 NT stores) for tiles above the diagonal, then
//      compute loop: 2 fp32 WMMAs, 16-lane suffix scan in the WMMA D layout,
//      carry from the right, NT b32 stores. 805 MB written exactly once,
//      non-temporal so the stream never evicts the 3 MB q/k from L2.

typedef __attribute__((ext_vector_type(2))) float v2f;
typedef __attribute__((ext_vector_type(4))) float v4f;
typedef __attribute__((ext_vector_type(8))) float v8f;

#define NTOK   4096
#define KDIM   1024
#define NHEAD  12
#define DHEAD  8
#define NPROJ  (2 * NHEAD * DHEAD)        // 192
#define QK_SCALE 0.35355339059327373f     // 8^-0.5

// ---------------------------------------------------------------------------
// Kernel 1: proj = x @ W^T, split/rearranged into qbuf/kbuf [h][n][8]
// One wave per 16x16 output tile; K=1024 in steps of 4 (fp32 WMMA).
// ---------------------------------------------------------------------------
__global__ __launch_bounds__(32)
void proj_qk_kernel(const float* __restrict__ x, const float* __restrict__ W,
                    float* __restrict__ qbuf, float* __restrict__ kbuf) {
  const int lane = threadIdx.x;
  const int hb   = lane >> 4;          // half-wave group (0/1)
  const int ln   = lane & 15;
  const int i0   = blockIdx.x * 16;    // row tile in [0,4096)
  const int o0   = blockIdx.y * 16;    // col tile in [0,192)

  // A: x[i0+ln][kk + v + 2*hb]   B: (W^T)[kk+.][o0+ln] = W[o0+ln][kk+.]
  const float* xrow = x + (size_t)(i0 + ln) * KDIM + 2 * hb;
  const float* wrow = W + (size_t)(o0 + ln) * KDIM + 2 * hb;

  v8f c = {};
#pragma unroll 4
  for (int kk = 0; kk < KDIM; kk += 4) {
    v2f a = *(const v2f*)(xrow + kk);
    v2f b = *(const v2f*)(wrow + kk);
    c = __builtin_amdgcn_wmma_f32_16x16x4_f32(false, a, false, b,
                                              (short)0, c, false, false);
  }

  // D: row = i0 + v + 8*hb, col o = o0 + ln (uniform over v for a lane).
  const int o = o0 + ln;
  float* dst;
  int d;
  if (o < NHEAD * DHEAD) { dst = qbuf; d = o; }
  else                   { dst = kbuf; d = o - NHEAD * DHEAD; }
  const int h = d / DHEAD;
  const int dd = d % DHEAD;
  float* p = dst + (size_t)h * NTOK * DHEAD + (size_t)(i0 + 8 * hb) * DHEAD + dd;
#pragma unroll
  for (int v = 0; v < 8; ++v) {
    p[v * DHEAD] = c[v];
  }
}

// ---------------------------------------------------------------------------
// Kernel 2: fused gates + log_sigmoid + causal mask + reverse cumsum + store.
// One wave per (head, 16-row band). Right-to-left tile sweep with carry.
// ---------------------------------------------------------------------------
__global__ __launch_bounds__(32)
void gates_scan_kernel(const float* __restrict__ qbuf,
                       const float* __restrict__ kbuf,
                       float* __restrict__ out) {
  const int lane = threadIdx.x;
  const int hb   = lane >> 4;
  const int ln   = lane & 15;
  const int tb   = blockIdx.x;         // diagonal tile index for this band
  const int i0   = tb * 16;            // row band
  const int h    = blockIdx.y;         // head

  const float* qh = qbuf + (size_t)h * NTOK * DHEAD;
  const float* kh = kbuf + (size_t)h * NTOK * DHEAD;

  // ---- Phase 1: tiles strictly above the diagonal band -> pure zero-fill.
  // Re-map lanes for wide stores: lane owns row (i0+ln), quads {hb, hb+2}.
  {
    float* pz = out + ((size_t)h * NTOK + i0 + ln) * NTOK + 4 * hb
              + (size_t)(NTOK - 16);
    const v4f z4 = {};
    for (int t = (NTOK / 16) - 1; t > tb; --t) {
      __builtin_nontemporal_store(z4, (v4f*)pz);
      __builtin_nontemporal_store(z4, (v4f*)(pz + 8));
      pz -= 16;
    }
  }

  // ---- Phase 2: compute sweep from the diagonal tile leftwards.
  // A operand (16 rows x K=8 split into two K=4 chunks), scale folded in.
  v2f a_lo = *(const v2f*)(qh + (size_t)(i0 + ln) * DHEAD + 2 * hb);
  v2f a_hi = *(const v2f*)(qh + (size_t)(i0 + ln) * DHEAD + 4 + 2 * hb);
  a_lo *= QK_SCALE;
  a_hi *= QK_SCALE;

  v8f carry = {};  // per-row running suffix sum from tiles to the right

  // Loop-carried pointers: one 64-bit decrement per pointer per tile;
  // the per-v / +4 deltas fold into instruction immediates.
  const float* kp = kh + (size_t)(16 * tb + ln) * DHEAD + 2 * hb;
  float* po = out + ((size_t)h * NTOK + i0 + 8 * hb) * NTOK + ln
            + (size_t)(16 * tb);

  for (int t = tb; t >= 0; --t) {
    const int j0  = t * 16;
    const int col = j0 + ln;

    v2f b_lo = *(const v2f*)(kp);
    v2f b_hi = *(const v2f*)(kp + 4);

    v8f cacc = {};
    cacc = __builtin_amdgcn_wmma_f32_16x16x4_f32(false, a_lo, false, b_lo,
                                                 (short)0, cacc, false, false);
    cacc = __builtin_amdgcn_wmma_f32_16x16x4_f32(false, a_hi, false, b_hi,
                                                 (short)0, cacc, false, false);

#pragma unroll
    for (int v = 0; v < 8; ++v) {
      const int row = i0 + v + 8 * hb;
      float z = cacc[v];
      // numerically stable log_sigmoid
      float g = fminf(z, 0.0f) - __logf(1.0f + __expf(-fabsf(z)));
      g = (col < row) ? g : 0.0f;                 // strict lower triangle

      // suffix (reverse) inclusive scan across the 16-lane half group:
      // lane ln ends with sum over cols >= ln within this tile.
#pragma unroll
      for (int off = 1; off < 16; off <<= 1) {
        float up = __shfl(g, lane + off, 32);
        if (ln + off < 16) g += up;
      }

      float r = g + carry[v];
      __builtin_nontemporal_store(r, po + (size_t)v * NTOK);
      // new carry = suffix sum at the leftmost column of this tile
      carry[v] = __shfl(r, hb * 16, 32);
    }

    kp -= 16 * DHEAD;
    po -= 16;
  }
}

// ---------------------------------------------------------------------------
extern "C" void kernel_launch(void* const* d_in, const int* in_sizes, int n_in,
                              void* d_out, int out_size, void* d_ws, size_t ws_size,
                              hipStream_t stream) {
  const float* x = (const float*)d_in[0];   // [1,4096,1024] fp32
  const float* W = (const float*)d_in[1];   // [192,1024]    fp32

  float* qbuf = (float*)d_ws;                       // [12][4096][8]
  float* kbuf = qbuf + (size_t)NHEAD * NTOK * DHEAD;

  proj_qk_kernel<<<dim3(NTOK / 16, NPROJ / 16), 32, 0, stream>>>(x, W, qbuf, kbuf);
  gates_scan_kernel<<<dim3(NTOK / 16, NHEAD), 32, 0, stream>>>(
      qbuf, kbuf, (float*)d_out);
}